// SoftMoe_41704132444187
// MI455X (gfx1250) — compile-verified
//
#include <hip/hip_runtime.h>
#include <math.h>

// ---------------------------------------------------------------------------
// Soft-MoE forward for MI455X (gfx1250), fp32 end-to-end via V_WMMA_F32_16X16X4_F32.
// Memory-bound (~400MB @ 23.3TB/s => ~17us floor): x, W1, W2 each read once;
// intermediates (logits 1MB, xs/h/ys <1MB) live in workspace and stay L2-hot.
// Multi-tile waves give independent load->WMMA chains for latency hiding.
// ---------------------------------------------------------------------------

typedef __attribute__((ext_vector_type(2))) float v2f;
typedef __attribute__((ext_vector_type(8))) float v8f;

#define BB 4
#define SS 4096
#define DD 768
#define EE 16
#define HH 3072
#define NSLOT 16   // E * SLOTS

static __device__ __forceinline__ v8f wmma4(v2f a, v2f b, v8f c) {
  // D = A(16x4 f32) * B(4x16 f32) + C(16x16 f32)
  return __builtin_amdgcn_wmma_f32_16x16x4_f32(
      /*neg_a=*/false, a, /*neg_b=*/false, b,
      /*c_mod=*/(short)0, c, /*reuse_a=*/false, /*reuse_b=*/false);
}

// -------------------- Kernel 1: logits + dispatch softmax -------------------
// One wave per 16-token tile. A = x[16 tok x 4 d], B = phi[4 d x 16 slot].
// C layout: VGPR j holds token (j + 8*half), slot = lane&15  ->  row softmax
// over the 16 slots = cross-lane reduction within each 16-lane half.
__global__ void k_logits(const float* __restrict__ x, const float* __restrict__ phi,
                         float* __restrict__ logitsT, float* __restrict__ disp) {
  int wave = blockIdx.x * (blockDim.x >> 5) + (threadIdx.x >> 5);
  int lane = threadIdx.x & 31;
  int half = lane >> 4, ln = lane & 15;
  int b  = wave / (SS / 16);
  int s0 = (wave % (SS / 16)) * 16;

  const float* arow = x + ((size_t)b * SS + s0 + ln) * DD;  // A row: token = ln
  v8f acc = {};
  for (int k = 0; k < DD; k += 4) {
    v2f a = *(const v2f*)(arow + k + 2 * half);             // K = k+2*half, +1
    v2f bf;
    bf.x = phi[(k + 2 * half) * NSLOT + ln];                // B col: slot = ln
    bf.y = phi[(k + 2 * half + 1) * NSLOT + ln];
    acc = wmma4(a, bf, acc);
  }

#pragma unroll
  for (int j = 0; j < 8; ++j) {
    float v = acc[j];
    float m = v;
#pragma unroll
    for (int d = 1; d < 16; d <<= 1) m = fmaxf(m, __shfl_xor(m, d, 16));
    float e = expf(v - m);
    float s = e;
#pragma unroll
    for (int d = 1; d < 16; d <<= 1) s += __shfl_xor(s, d, 16);
    int tok = s0 + j + 8 * half;
    logitsT[((size_t)b * NSLOT + ln) * SS + tok] = v;       // [b][n][s] for col access
    disp[((size_t)b * SS + tok) * NSLOT + ln] = e / s;      // dispatch softmax (over n)
  }
}

// -------------------- Kernel 2: combine-softmax column stats ----------------
__global__ void k_colstats(const float* __restrict__ logitsT,
                           float* __restrict__ cmax, float* __restrict__ csum) {
  __shared__ float red[256];
  int bn = blockIdx.x;                 // b*16 + n, 64 total
  int tid = threadIdx.x;
  const float* col = logitsT + (size_t)bn * SS;
  float m = -INFINITY;
  for (int s = tid; s < SS; s += 256) m = fmaxf(m, col[s]);
  red[tid] = m; __syncthreads();
  for (int o = 128; o; o >>= 1) { if (tid < o) red[tid] = fmaxf(red[tid], red[tid + o]); __syncthreads(); }
  m = red[0]; __syncthreads();
  float s = 0.f;
  for (int i = tid; i < SS; i += 256) s += expf(col[i] - m);
  red[tid] = s; __syncthreads();
  for (int o = 128; o; o >>= 1) { if (tid < o) red[tid] += red[tid + o]; __syncthreads(); }
  if (tid == 0) { cmax[bn] = m; csum[bn] = red[0]; }
}

__global__ void k_zero(float* __restrict__ p, int n) {
  int i = blockIdx.x * blockDim.x + threadIdx.x;
  if (i < n) p[i] = 0.f;
}

// -------------------- Kernel 3: xs = dispatch^T @ x -------------------------
// Per wave: one (b, d-tile, K-chunk). A = dispatch^T[16 slot x 4 seq],
// B = x[4 seq x 16 d]. S split into 16 chunks, accumulate with f32 atomics.
__global__ void k_xs(const float* __restrict__ x, const float* __restrict__ disp,
                     float* __restrict__ xs) {
  int wave = blockIdx.x * (blockDim.x >> 5) + (threadIdx.x >> 5);
  int lane = threadIdx.x & 31;
  int half = lane >> 4, ln = lane & 15;
  int chunk = wave & 15;
  int dt = (wave >> 4) % (DD / 16);
  int b  = wave / (16 * (DD / 16));
  int d0 = dt * 16;
  int sbeg = chunk * (SS / 16);

  v8f acc = {};
  for (int s = sbeg; s < sbeg + SS / 16; s += 4) {
    int r0 = s + 2 * half;
    v2f a, bf;
    a.x  = disp[((size_t)b * SS + r0) * NSLOT + ln];        // A row: slot = ln
    a.y  = disp[((size_t)b * SS + r0 + 1) * NSLOT + ln];
    bf.x = x[((size_t)b * SS + r0) * DD + d0 + ln];         // B col: d = ln
    bf.y = x[((size_t)b * SS + r0 + 1) * DD + d0 + ln];
    acc = wmma4(a, bf, acc);
  }
#pragma unroll
  for (int j = 0; j < 8; ++j) {
    int slot = j + 8 * half;
    atomicAdd(&xs[((size_t)b * NSLOT + slot) * DD + d0 + ln], acc[j]);
  }
}

// -------------------- Kernel 4: h = GELU(xs @ W1 + b1) ----------------------
// A rows = 4 batches (padded to 16 via clamp), K = D. Two 16-wide H tiles per
// wave share the A fragment -> 2 independent WMMA chains, half the A loads.
__global__ void k_ffn1(const float* __restrict__ xs, const float* __restrict__ W1,
                       const float* __restrict__ b1, float* __restrict__ h) {
  int wave = blockIdx.x * (blockDim.x >> 5) + (threadIdx.x >> 5);
  int lane = threadIdx.x & 31;
  int half = lane >> 4, ln = lane & 15;
  int e  = wave / (HH / 32);
  int h0 = (wave % (HH / 32)) * 32;

  const float* arow = xs + ((size_t)(ln & 3) * NSLOT + e) * DD;  // batch = row (clamped)
  const float* wb0  = W1 + (size_t)e * DD * HH + h0 + ln;
  const float* wb1  = wb0 + 16;
  v8f acc0 = {}, acc1 = {};
  for (int k = 0; k < DD; k += 4) {
    size_t r0 = (size_t)(k + 2 * half) * HH;
    v2f a = *(const v2f*)(arow + k + 2 * half);
    v2f bf0, bf1;
    bf0.x = wb0[r0];       bf0.y = wb0[r0 + HH];
    bf1.x = wb1[r0];       bf1.y = wb1[r0 + HH];
    __builtin_prefetch(wb0 + r0 + (size_t)8 * HH, 0, 0);
    acc0 = wmma4(a, bf0, acc0);
    acc1 = wmma4(a, bf1, acc1);
  }
  if (half == 0) {
#pragma unroll
    for (int j = 0; j < 4; ++j) {                // rows 0..3 = batches 0..3
      float v0 = acc0[j] + b1[e * HH + h0 + ln];
      float v1 = acc1[j] + b1[e * HH + h0 + 16 + ln];
      v0 = 0.5f * v0 * (1.0f + erff(v0 * 0.70710678118654752f));  // exact GELU
      v1 = 0.5f * v1 * (1.0f + erff(v1 * 0.70710678118654752f));
      h[((size_t)j * NSLOT + e) * HH + h0 + ln] = v0;
      h[((size_t)j * NSLOT + e) * HH + h0 + 16 + ln] = v1;
    }
  }
}

// -------------------- Kernel 5: ys += h @ W2 (K split 4x, atomics) ----------
// Two 16-wide D tiles per wave; kc==0 waves fold the b2 bias into their
// partial so no separate bias kernel is needed.
__global__ void k_ffn2(const float* __restrict__ h, const float* __restrict__ W2,
                       const float* __restrict__ b2, float* __restrict__ ys) {
  int wave = blockIdx.x * (blockDim.x >> 5) + (threadIdx.x >> 5);
  int lane = threadIdx.x & 31;
  int half = lane >> 4, ln = lane & 15;
  int kc  = wave & 3;
  int dt2 = (wave >> 2) % (DD / 32);
  int e   = wave / (4 * (DD / 32));
  int d0  = dt2 * 32;
  int kbeg = kc * (HH / 4);

  const float* arow = h + ((size_t)(ln & 3) * NSLOT + e) * HH;
  const float* wb0  = W2 + (size_t)e * HH * DD + d0 + ln;
  const float* wb1  = wb0 + 16;
  v8f acc0 = {}, acc1 = {};
  for (int k = kbeg; k < kbeg + HH / 4; k += 4) {
    size_t r0 = (size_t)(k + 2 * half) * DD;
    v2f a = *(const v2f*)(arow + k + 2 * half);
    v2f bf0, bf1;
    bf0.x = wb0[r0];       bf0.y = wb0[r0 + DD];
    bf1.x = wb1[r0];       bf1.y = wb1[r0 + DD];
    __builtin_prefetch(wb0 + r0 + (size_t)8 * DD, 0, 0);
    acc0 = wmma4(a, bf0, acc0);
    acc1 = wmma4(a, bf1, acc1);
  }
  if (half == 0) {
    float bias0 = (kc == 0) ? b2[e * DD + d0 + ln] : 0.0f;
    float bias1 = (kc == 0) ? b2[e * DD + d0 + 16 + ln] : 0.0f;
#pragma unroll
    for (int j = 0; j < 4; ++j) {
      atomicAdd(&ys[((size_t)j * NSLOT + e) * DD + d0 + ln], acc0[j] + bias0);
      atomicAdd(&ys[((size_t)j * NSLOT + e) * DD + d0 + 16 + ln], acc1[j] + bias1);
    }
  }
}

// -------------------- Kernel 6: y = combine @ ys ----------------------------
// A = combine weights [16 tok x 4 slot], built on the fly from logitsT +
// column stats. Four 16-wide D tiles per wave share one A fragment: 4x fewer
// expf/logits loads, 4 independent WMMA chains, 256B of y per row per wave.
__global__ void k_combine(const float* __restrict__ logitsT, const float* __restrict__ cmax,
                          const float* __restrict__ csum, const float* __restrict__ ys,
                          float* __restrict__ y) {
  int wave = blockIdx.x * (blockDim.x >> 5) + (threadIdx.x >> 5);
  int lane = threadIdx.x & 31;
  int half = lane >> 4, ln = lane & 15;
  int dt4 = wave % (DD / 64);
  int st  = (wave / (DD / 64)) % (SS / 16);
  int b   = wave / ((DD / 64) * (SS / 16));
  int s0 = st * 16, d0 = dt4 * 64;

  v8f acc[4] = {{}, {}, {}, {}};
#pragma unroll
  for (int k = 0; k < NSLOT; k += 4) {
    int n0 = k + 2 * half;
    float cm0 = cmax[b * NSLOT + n0],     cs0 = csum[b * NSLOT + n0];
    float cm1 = cmax[b * NSLOT + n0 + 1], cs1 = csum[b * NSLOT + n0 + 1];
    v2f a;
    a.x = expf(logitsT[((size_t)b * NSLOT + n0) * SS + s0 + ln] - cm0) / cs0;
    a.y = expf(logitsT[((size_t)b * NSLOT + n0 + 1) * SS + s0 + ln] - cm1) / cs1;
    const float* yr0 = ys + ((size_t)b * NSLOT + n0) * DD + d0 + ln;
#pragma unroll
    for (int t = 0; t < 4; ++t) {
      v2f bf;
      bf.x = yr0[16 * t];
      bf.y = yr0[16 * t + DD];
      acc[t] = wmma4(a, bf, acc[t]);
    }
  }
#pragma unroll
  for (int j = 0; j < 8; ++j) {
    int tok = s0 + j + 8 * half;
    float* yrow = y + ((size_t)b * SS + tok) * DD + d0 + ln;
#pragma unroll
    for (int t = 0; t < 4; ++t) yrow[16 * t] = acc[t][j];
  }
}

// ---------------------------------------------------------------------------
extern "C" void kernel_launch(void* const* d_in, const int* in_sizes, int n_in,
                              void* d_out, int out_size, void* d_ws, size_t ws_size,
                              hipStream_t stream) {
  const float* x   = (const float*)d_in[0];
  const float* phi = (const float*)d_in[1];
  const float* W1  = (const float*)d_in[2];
  const float* b1  = (const float*)d_in[3];
  const float* W2  = (const float*)d_in[4];
  const float* b2  = (const float*)d_in[5];
  float* y  = (float*)d_out;
  float* ws = (float*)d_ws;

  // Workspace layout (floats): logitsT | disp | cmax | csum | xs | ys | h
  float* logitsT = ws;                           // B*S*16   = 262144
  float* disp    = logitsT + BB * SS * NSLOT;    // B*S*16   = 262144
  float* cmaxp   = disp + BB * SS * NSLOT;       // 64
  float* csump   = cmaxp + BB * NSLOT;           // 64
  float* xs      = csump + BB * NSLOT;           // B*16*D   = 49152
  float* ysb     = xs + BB * NSLOT * DD;         // B*16*D   = 49152
  float* hbuf    = ysb + BB * NSLOT * DD;        // B*16*H   = 196608
  (void)in_sizes; (void)n_in; (void)out_size; (void)ws_size;

  // 1) logits + dispatch softmax: B*S/16 = 1024 waves
  k_logits<<<1024 / 8, 256, 0, stream>>>(x, phi, logitsT, disp);
  // 2) combine-softmax column stats: 64 (b,n) reductions
  k_colstats<<<BB * NSLOT, 256, 0, stream>>>(logitsT, cmaxp, csump);
  // 3) zero xs + ys (contiguous), then xs accumulation: 3072 waves
  k_zero<<<(2 * BB * NSLOT * DD + 255) / 256, 256, 0, stream>>>(xs, 2 * BB * NSLOT * DD);
  k_xs<<<(BB * (DD / 16) * 16) / 8, 256, 0, stream>>>(x, disp, xs);
  // 4) expert FFN layer 1 (GELU): 16*96 = 1536 waves, 2 H-tiles each
  k_ffn1<<<(EE * (HH / 32)) / 8, 256, 0, stream>>>(xs, W1, b1, hbuf);
  // 5) expert FFN layer 2 (+bias fold): 16*24*4 = 1536 waves, 2 D-tiles each
  k_ffn2<<<(EE * (DD / 32) * 4) / 8, 256, 0, stream>>>(hbuf, W2, b2, ysb);
  // 6) combine: 4*256*12 = 12288 waves, 4 D-tiles each
  k_combine<<<(BB * (SS / 16) * (DD / 64)) / 8, 256, 0, stream>>>(logitsT, cmaxp, csump, ysb, y);
}